// IcoRePaConv_8641474199776
// MI455X (gfx1250) — compile-verified
//
#include <hip/hip_runtime.h>

typedef __attribute__((ext_vector_type(2))) float v2f;
typedef __attribute__((ext_vector_type(8))) float v8f;

#define BB 2
#define CC 32
#define NN 40962
#define KK 25
#define OO 64
#define CK 800       // CC*KK, GEMM reduction dim
#define FEAT_LD 804  // padded LDS row stride (floats): even (8B-aligned b64 loads),
                     // 804%64=36, gcd(36,64)=4 -> 16 rows hit 16 distinct banks
#define TROWS 16     // points (n) per block = WMMA M

__global__ __launch_bounds__(128) void icorepa_fused_kernel(
    const float* __restrict__ x,        // [B][C][N]
    const float* __restrict__ nw,       // [N][K][3]
    const float* __restrict__ W,        // [O][C*K]
    const float* __restrict__ bias,     // [O]
    const int*   __restrict__ nidx,     // [N][K][3]
    float*       __restrict__ out)      // [B][O][N]
{
    __shared__ float s_feat[TROWS * FEAT_LD];   // 51456 B
    __shared__ int   s_idx [TROWS * KK * 3];    //  4800 B
    __shared__ float s_w   [TROWS * KK * 3];    //  4800 B
    __shared__ float s_out [OO * TROWS];        //  4096 B

    const int tid    = threadIdx.x;
    const int n_base = blockIdx.x * TROWS;
    const int bb     = blockIdx.y;

    // ---- Phase 0: stage neighbor indices & weights for 16 points ----
    for (int u = tid; u < TROWS * KK * 3; u += 128) {
        int row = u / (KK * 3);
        int j   = u - row * (KK * 3);
        int n   = n_base + row; if (n >= NN) n = NN - 1;   // clamp edge tile
        s_idx[u] = nidx[(size_t)n * (KK * 3) + j];
        s_w[u]   = nw  [(size_t)n * (KK * 3) + j];
    }
    __syncthreads();

    // ---- Phase 1: gather + 3-tap einsum -> feat[row][c*25+k] in LDS ----
    const float* xb = x + (size_t)bb * CC * NN;
    for (int u = tid; u < TROWS * CK; u += 128) {
        int row = u / CK;
        int ck  = u - row * CK;
        int c   = ck / KK;
        int k   = ck - c * KK;
        int base = row * (KK * 3) + k * 3;
        int i0 = s_idx[base + 0], i1 = s_idx[base + 1], i2 = s_idx[base + 2];
        float w0 = s_w[base + 0], w1 = s_w[base + 1], w2 = s_w[base + 2];
        const float* xc = xb + (size_t)c * NN;
        float s = fmaf(xc[i0], w0, fmaf(xc[i1], w1, xc[i2] * w2));
        s_feat[row * FEAT_LD + ck] = s;
    }
    __syncthreads();

    // ---- Phase 2: tall-skinny GEMM via V_WMMA_F32_16X16X4_F32 ----
    // Each wave computes D[16 rows][16 outputs]; o_base = wave*16.
    const int lane = tid & 31;          // wave32
    const int wave = tid >> 5;
    const int hi   = lane >> 4;         // lane half selects K-pair {0,1} vs {2,3}
    const int l15  = lane & 15;
    const int o    = wave * 16 + l15;   // this lane's output column

    v8f acc = {};
    // A (16x4 f32): M = lane&15 both halves; VGPR0 = K+0 / K+2, VGPR1 = K+1 / K+3
    const float* aptr = &s_feat[l15 * FEAT_LD + 2 * hi];
    // B (4x16 f32): N-col = lane&15; VGPR0 = K+0 / K+2 rows, VGPR1 = K+1 / K+3
    const float* bptr = W + (size_t)o * CK + 2 * hi;

    #pragma unroll 4
    for (int kk = 0; kk < CK; kk += 4) {
        v2f a  = *(const v2f*)(aptr + kk);   // ds_load_b64 (8B aligned)
        v2f bv = *(const v2f*)(bptr + kk);   // global_load_b64 (L2-resident W)
        acc = __builtin_amdgcn_wmma_f32_16x16x4_f32(
                  /*neg_a=*/false, a, /*neg_b=*/false, bv,
                  /*c_mod=*/(short)0, acc, /*reuse_a=*/false, /*reuse_b=*/false);
    }

    // D layout: acc[i] = element (M = i + 8*hi, Ncol = lane&15).
    // Transpose through LDS so stores are contiguous in n.
    #pragma unroll
    for (int i = 0; i < 8; ++i) {
        s_out[o * TROWS + (i + 8 * hi)] = acc[i];
    }
    __syncthreads();

    // ---- Phase 3: coalesced store + bias: 16 lanes write 16 consecutive n ----
    for (int u = tid; u < OO * TROWS; u += 128) {
        int oo = u >> 4;
        int m  = u & 15;
        int n  = n_base + m;
        if (n < NN)
            out[((size_t)bb * OO + oo) * NN + n] = s_out[u] + bias[oo];
    }
}

extern "C" void kernel_launch(void* const* d_in, const int* in_sizes, int n_in,
                              void* d_out, int out_size, void* d_ws, size_t ws_size,
                              hipStream_t stream) {
    (void)in_sizes; (void)n_in; (void)d_ws; (void)ws_size; (void)out_size;
    const float* x    = (const float*)d_in[0];   // (2, 32, 40962)
    const float* nw   = (const float*)d_in[1];   // (40962, 25, 3)
    const float* W    = (const float*)d_in[2];   // (64, 800)
    const float* bias = (const float*)d_in[3];   // (64,)
    const int*   nidx = (const int*)  d_in[4];   // (40962, 25, 3)
    float*       out  = (float*)d_out;           // (2, 64, 40962)

    dim3 grid((NN + TROWS - 1) / TROWS, BB);     // (2561, 2)
    dim3 block(128);                              // 4 wave32 waves
    icorepa_fused_kernel<<<grid, block, 0, stream>>>(x, nw, W, bias, nidx, out);
}